// Gaterouter_47201690583342
// MI455X (gfx1250) — compile-verified
//
#include <hip/hip_runtime.h>
#include <math.h>

typedef __attribute__((ext_vector_type(2))) float v2f;
typedef __attribute__((ext_vector_type(8))) float v8f;

#define NROWS 16384
#define DIM   4096
#define NEXP  64
#define TOPK  8

#define ROWS_PB   128          // rows per block: 8 waves x 16 rows
#define CK        64           // K-chunk staged in LDS per double-buffer slot
#define XS_STRIDE 68           // 64 data dwords + 4 pad -> stride==4 (mod 64):
                               // conflict-free ds_load_b64 for the WMMA A fetch

// ---------------------------------------------------------------------------
// gfx1250 async-to-LDS DMA (ASYNCcnt-tracked). The LDS address operand is the
// byte address within the wave's LDS allocation (static __shared__ block
// starts at offset 0). IMPORTANT: the generic pointer to the LDS destination
// is passed as an extra asm input so the shared object ESCAPES — otherwise
// LLVM proves the LDS array is never written and deletes the consumer loads.
// ---------------------------------------------------------------------------
__device__ __forceinline__ void async_copy_b128(unsigned lds_byte_off,
                                                const float* gsrc,
                                                const float* ldst_escape) {
    asm volatile("global_load_async_to_lds_b128 %0, %1, off"
                 :: "v"(lds_byte_off), "v"(gsrc), "v"(ldst_escape)
                 : "memory");
}
__device__ __forceinline__ void wait_async0() {
    asm volatile("s_wait_asynccnt 0x0" ::: "memory");
}

// ---------------------------------------------------------------------------
// Kernel 1: logits = x @ W^T via V_WMMA_F32_16X16X4_F32.
// Block = 256 threads = 8 waves; each wave owns 16 rows x 64 experts
// (four 16x16 f32 accumulators). x is double-buffered in LDS via async DMA;
// W (1 MB, L2-resident) is read directly from global on the VMEM pipe.
// ---------------------------------------------------------------------------
__global__ __launch_bounds__(256) void gate_gemm_wmma(
    const float* __restrict__ x,
    const float* __restrict__ W,
    float* __restrict__ logits)
{
    __shared__ float xs[2][ROWS_PB * XS_STRIDE];

    const int tid      = threadIdx.x;
    const int lane     = tid & 31;
    const int wave     = tid >> 5;
    const int blockRow = blockIdx.x * ROWS_PB;
    const int rowBase  = blockRow + wave * 16;

    const int m    = lane & 15;          // A: row in tile / B: expert in tile
    const int koff = (lane >> 4) << 1;   // lanes 16..31 carry K=2,3 of each step

    // W pointers for the 4 expert tiles (B operand: lane holds W[n][k..k+1])
    const float* w0 = W + (size_t)( 0 + m) * DIM + koff;
    const float* w1 = W + (size_t)(16 + m) * DIM + koff;
    const float* w2 = W + (size_t)(32 + m) * DIM + koff;
    const float* w3 = W + (size_t)(48 + m) * DIM + koff;

    v8f c0 = {}, c1 = {}, c2 = {}, c3 = {};

    // Stage one 128-row x CK-col tile of x into xs[buf] (2048 x b128, 8/thread)
    auto stage = [&](int buf, int kb) {
#pragma unroll
        for (int j = 0; j < (ROWS_PB * (CK / 4)) / 256; ++j) {
            const int ci  = tid + 256 * j;
            const int row = ci >> 4;         // 0..127
            const int c4  = (ci & 15) * 4;   // 0,4,...,60
            const float* gsrc = x + (size_t)(blockRow + row) * DIM + kb + c4;
            const float* ldst = &xs[buf][row * XS_STRIDE + c4];
            const unsigned off =
                (unsigned)((const char*)ldst - (const char*)&xs[0][0]);
            async_copy_b128(off, gsrc, ldst);
        }
    };

    stage(0, 0);

    for (int kb = 0; kb < DIM; kb += CK) {
        const int cur = (kb / CK) & 1;

        wait_async0();        // my DMA into xs[cur] is complete
        __syncthreads();      // everyone's DMA done + nobody still reads xs[cur^1]

        if (kb + CK < DIM) stage(cur ^ 1, kb + CK);  // prefetch next chunk

        const float* xsrow = &xs[cur][(wave * 16 + m) * XS_STRIDE + koff];
#pragma unroll
        for (int kc = 0; kc < CK; kc += 4) {
            v2f a  = *(const v2f*)(xsrow + kc);        // ds_load_b64, no conflicts
            v2f b0 = *(const v2f*)(w0 + kb + kc);
            v2f b1 = *(const v2f*)(w1 + kb + kc);
            v2f b2 = *(const v2f*)(w2 + kb + kc);
            v2f b3 = *(const v2f*)(w3 + kb + kc);
            // 8 args: (neg_a, A, neg_b, B, c_mod, C, reuse_a, reuse_b)
            c0 = __builtin_amdgcn_wmma_f32_16x16x4_f32(false, a, false, b0,
                                                       (short)0, c0, false, false);
            c1 = __builtin_amdgcn_wmma_f32_16x16x4_f32(false, a, false, b1,
                                                       (short)0, c1, false, false);
            c2 = __builtin_amdgcn_wmma_f32_16x16x4_f32(false, a, false, b2,
                                                       (short)0, c2, false, false);
            c3 = __builtin_amdgcn_wmma_f32_16x16x4_f32(false, a, false, b3,
                                                       (short)0, c3, false, false);
        }
    }

    // C/D layout: VGPR i <-> M = i + 8*(lane>=16), N = lane&15
    float* outBase = logits + (size_t)rowBase * NEXP;
    const int mAdd = (lane >> 4) * 8;
    const int n    = lane & 15;
#pragma unroll
    for (int i = 0; i < 8; ++i) {
        const size_t r = (size_t)(i + mAdd) * NEXP;
        outBase[r +  0 + n] = c0[i];
        outBase[r + 16 + n] = c1[i];
        outBase[r + 32 + n] = c2[i];
        outBase[r + 48 + n] = c3[i];
    }
}

// ---------------------------------------------------------------------------
// Kernel 2: per-row top-8 (lax.top_k semantics: descending, lowest index on
// ties) + sparse softmax. One thread per row; values stay in registers by
// re-scanning with an exclusion predicate (no dynamic register indexing).
// Operates in place over the logits region (reads full row before writing).
// ---------------------------------------------------------------------------
__global__ __launch_bounds__(256) void gate_topk_softmax(
    float* __restrict__ probs,       // [NROWS, NEXP] in-place (holds logits)
    const float* __restrict__ bias,  // [NEXP]
    int* __restrict__ out_idx)       // [NROWS, TOPK]
{
    const int row = blockIdx.x * blockDim.x + threadIdx.x;
    if (row >= NROWS) return;

    float* rowp = probs + (size_t)row * NEXP;

    float v[NEXP];
#pragma unroll
    for (int e = 0; e < NEXP; ++e) v[e] = rowp[e] + bias[e];

    float tv[TOPK];
    int   ti[TOPK];
    float prevV = INFINITY;
    int   prevI = -1;
#pragma unroll
    for (int kkk = 0; kkk < TOPK; ++kkk) {
        float best = -INFINITY;
        int   bi   = 0;
#pragma unroll
        for (int e = 0; e < NEXP; ++e) {
            const bool elig = (v[e] < prevV) || ((v[e] == prevV) && (e > prevI));
            if (elig && (v[e] > best)) { best = v[e]; bi = e; }
        }
        tv[kkk] = best;
        ti[kkk] = bi;
        prevV = best;
        prevI = bi;
    }

    const float mx = tv[0];
    float ev[TOPK];
    float sum = 0.0f;
#pragma unroll
    for (int kkk = 0; kkk < TOPK; ++kkk) {
        ev[kkk] = __expf(tv[kkk] - mx);
        sum += ev[kkk];
    }
    const float inv = 1.0f / sum;

#pragma unroll
    for (int e = 0; e < NEXP; ++e) rowp[e] = 0.0f;
#pragma unroll
    for (int kkk = 0; kkk < TOPK; ++kkk) rowp[ti[kkk]] = ev[kkk] * inv;

    int* irow = out_idx + (size_t)row * TOPK;
#pragma unroll
    for (int kkk = 0; kkk < TOPK; ++kkk) irow[kkk] = ti[kkk];
}

// ---------------------------------------------------------------------------
extern "C" void kernel_launch(void* const* d_in, const int* in_sizes, int n_in,
                              void* d_out, int out_size, void* d_ws, size_t ws_size,
                              hipStream_t stream) {
    (void)in_sizes; (void)n_in; (void)out_size; (void)d_ws; (void)ws_size;

    const float* x = (const float*)d_in[0];   // [16384, 4096]
    const float* W = (const float*)d_in[1];   // [64, 4096]
    const float* b = (const float*)d_in[2];   // [64]

    float* probs = (float*)d_out;                                  // [16384,64]
    int*   idx   = (int*)((float*)d_out + (size_t)NROWS * NEXP);   // [16384,8]

    // GEMM: 128 blocks x 8 waves; each wave owns 16 rows x 64 experts.
    gate_gemm_wmma<<<NROWS / ROWS_PB, 256, 0, stream>>>(x, W, probs);

    // Top-k + softmax: one thread per row.
    gate_topk_softmax<<<NROWS / 256, 256, 0, stream>>>(probs, b, idx);
}